// Repro_11227044512409
// MI455X (gfx1250) — compile-verified
//
#include <hip/hip_runtime.h>
#include <hip/hip_bf16.h>
#include <stdint.h>

// Cross-entropy backward, streaming bandwidth kernel for gfx1250 (MI455X).
// grad[i,j] = exp(logp[i,j]) * g_i - (j==tgt_i) * g_i,  g_i = 0 for ignore rows.
//
// Memory-bound: ~2.1 GB traffic -> ~90us floor at 23.3 TB/s. Uses CDNA5
// async-to-LDS staging (ASYNCcnt) double-buffered per wave, NT cache hints
// on both the streamed input and the streamed output.

#define IGNORE_INDEX (-100)
#define TILE_ELEMS 2048   // 256 threads * 8 bf16 per thread
#define TILE_UINTS (TILE_ELEMS / 2)

typedef unsigned int u32x4 __attribute__((ext_vector_type(4)));

__device__ __forceinline__ unsigned int pack_bf16x2_rne(float lo, float hi) {
  unsigned int a = __float_as_uint(lo);
  unsigned int b = __float_as_uint(hi);
  a += 0x7FFFu + ((a >> 16) & 1u);   // round-to-nearest-even bf16
  b += 0x7FFFu + ((b >> 16) & 1u);
  return (a >> 16) | (b & 0xFFFF0000u);
}

__global__ __launch_bounds__(256) void ce_backward_kernel(
    const __hip_bfloat16* __restrict__ logp,   // [N, V] saved log-softmax
    const int* __restrict__ tgt,               // [N] target ids
    const __hip_bfloat16* __restrict__ dy,     // [N] upstream grads
    __hip_bfloat16* __restrict__ grad,         // [N, V] output
    int V)
{
  __shared__ unsigned int smem[2][TILE_UINTS]; // 2 x 4KB staging buffers

  const int row = blockIdx.y;
  const long long rowBase = (long long)row * (long long)V;

  // Uniform per-row scalars (row is block-uniform -> scalar path).
  const int t = tgt[row];
  float g = 0.0f;
  int tcol = -1;
  if (t != IGNORE_INDEX) {
    g = __bfloat162float(dy[row]);
    tcol = t;
  }

  const unsigned int laneU = threadIdx.x * 4u;          // uint index: 16B/lane
  unsigned int ldsOff[2];
  // Low 32 bits of the flat address of an LDS object == DS byte offset.
  ldsOff[0] = (unsigned int)(uintptr_t)&smem[0][laneU];
  ldsOff[1] = (unsigned int)(uintptr_t)&smem[1][laneU];

  const int tilesPerRow = (V + TILE_ELEMS - 1) / TILE_ELEMS;   // 63 for V=128256

  int tile = blockIdx.x;
  if (tile >= tilesPerRow) return;

  // Prologue: stage first tile into buffer 0 (per-lane 16B async copy).
  {
    const int c = tile * TILE_ELEMS + (int)threadIdx.x * 8;
    if (c < V) {
      unsigned long long ga =
          (unsigned long long)(uintptr_t)(logp + rowBase + c);
      asm volatile("global_load_async_to_lds_b128 %0, %1, off th:TH_LOAD_NT"
                   :: "v"(ldsOff[0]), "v"(ga) : "memory");
    }
  }

  int buf = 0;
  for (; tile < tilesPerRow; tile += gridDim.x) {
    const int nextTile = tile + (int)gridDim.x;   // block-uniform condition
    if (nextTile < tilesPerRow) {
      // Kick off next tile into the other buffer, then wait only for the
      // current tile (async loads complete in order -> cnt<=1 is enough).
      const int cn = nextTile * TILE_ELEMS + (int)threadIdx.x * 8;
      if (cn < V) {
        unsigned long long ga =
            (unsigned long long)(uintptr_t)(logp + rowBase + cn);
        asm volatile("global_load_async_to_lds_b128 %0, %1, off th:TH_LOAD_NT"
                     :: "v"(ldsOff[buf ^ 1]), "v"(ga) : "memory");
      }
      asm volatile("s_wait_asynccnt 0x1" ::: "memory");
    } else {
      asm volatile("s_wait_asynccnt 0x0" ::: "memory");
    }

    const int c = tile * TILE_ELEMS + (int)threadIdx.x * 8;
    if (c < V) {   // V % 8 == 0: chunk is all-in or all-out
      const u32x4 p = *reinterpret_cast<const u32x4*>(&smem[buf][laneU]);

      float f[8];
      f[0] = __uint_as_float(p.x << 16);
      f[1] = __uint_as_float(p.x & 0xFFFF0000u);
      f[2] = __uint_as_float(p.y << 16);
      f[3] = __uint_as_float(p.y & 0xFFFF0000u);
      f[4] = __uint_as_float(p.z << 16);
      f[5] = __uint_as_float(p.z & 0xFFFF0000u);
      f[6] = __uint_as_float(p.w << 16);
      f[7] = __uint_as_float(p.w & 0xFFFF0000u);

      float r[8];
#pragma unroll
      for (int j = 0; j < 8; ++j) r[j] = __expf(f[j]) * g;

      // one-hot subtraction (8 cmp+cndmask, no indexed array -> stays in VGPRs)
#pragma unroll
      for (int j = 0; j < 8; ++j)
        if (c + j == tcol) r[j] -= g;

      u32x4 o;
      o.x = pack_bf16x2_rne(r[0], r[1]);
      o.y = pack_bf16x2_rne(r[2], r[3]);
      o.z = pack_bf16x2_rne(r[4], r[5]);
      o.w = pack_bf16x2_rne(r[6], r[7]);

      __builtin_nontemporal_store(
          o, reinterpret_cast<u32x4*>(grad + rowBase + c));
    }
    buf ^= 1;
  }
}

extern "C" void kernel_launch(void* const* d_in, const int* in_sizes, int n_in,
                              void* d_out, int out_size, void* d_ws, size_t ws_size,
                              hipStream_t stream) {
  // setup_inputs order: primals_2 (int targets, N), convert_element_type_1
  // (bf16 [N,V] log-softmax), tangents_1 (bf16 [N]).
  const int*            tgt  = (const int*)d_in[0];
  const __hip_bfloat16* logp = (const __hip_bfloat16*)d_in[1];
  const __hip_bfloat16* dy   = (const __hip_bfloat16*)d_in[2];
  __hip_bfloat16*       grad = (__hip_bfloat16*)d_out;

  const int N = in_sizes[0];
  const int V = (N > 0) ? (int)(in_sizes[1] / N) : 0;   // 128256

  // 16 blocks per row -> each block double-buffers across ~4 tiles of 4KB.
  dim3 grid(16, (unsigned)N);
  ce_backward_kernel<<<grid, 256, 0, stream>>>(logp, tgt, dy, grad, V);
}